// Convolution_40759239639120
// MI455X (gfx1250) — compile-verified
//
#include <hip/hip_runtime.h>
#include <hip/hip_bf16.h>
#include <math.h>

typedef __attribute__((ext_vector_type(2))) float v2f;
typedef __attribute__((ext_vector_type(4))) float v4f;
typedef __attribute__((ext_vector_type(8))) float v8f;

#define WAVES_PB    8
#define DIMF        40
#define INV_S3      0.57735026918962576f
// (1/8 W2-norm) * (1/sqrt(24) fan) * (1/4 neighbor-norm)
#define OUT_SCALE   (0.20412414523193154f * 0.03125f)
#define W2P_ELEMS   36864   // 36 j * 8 kp * 2 half * 16 n * 4 q
#define W1P_ELEMS   1024    // 4 nt * 2 kp * 2 half * 16 n * 4 q

__global__ void zero_out_kernel(float* __restrict__ out, int n) {
    int i = blockIdx.x * blockDim.x + threadIdx.x;
    if (i < n) out[i] = 0.0f;
}

// Repack W2/W1 into per-lane WMMA B-operand order so the hot loop uses b128
// loads: W2p[j][kp][half][n][q], q -> (kk = 2kp + (q>>1), t = q&1),
// value = W2[(4kk + 2half + t)*576 + 16j + n]. Same scheme for W1 (cols=64).
__global__ void repack_kernel(const float* __restrict__ W1,
                              const float* __restrict__ W2,
                              float* __restrict__ W2p,
                              float* __restrict__ W1p) {
    int i = blockIdx.x * blockDim.x + threadIdx.x;
    if (i < W2P_ELEMS) {
        const int q    = i & 3;
        const int n    = (i >> 2) & 15;
        const int half = (i >> 6) & 1;
        const int kp   = (i >> 7) & 7;
        const int j    = i >> 10;
        const int kk   = 2 * kp + (q >> 1);
        const int kr   = 4 * kk + 2 * half + (q & 1);
        W2p[i] = W2[kr * 576 + 16 * j + n];
    } else if (i < W2P_ELEMS + W1P_ELEMS) {
        const int p    = i - W2P_ELEMS;
        const int q    = p & 3;
        const int n    = (p >> 2) & 15;
        const int half = (p >> 6) & 1;
        const int kp   = (p >> 7) & 1;
        const int nt   = p >> 8;
        const int kk   = 2 * kp + (q >> 1);
        const int kr   = 4 * kk + 2 * half + (q & 1);
        W1p[p] = W1[kr * 64 + 16 * nt + n];
    }
}

// w-tile j of GEMM2: K=64 in 16 WMMA steps; B operands via 8 b128 loads.
__device__ __forceinline__ v8f gemm2_tile(const float* __restrict__ W2p,
                                          const v2f* a2, int j, int laneoff) {
    v8f acc = {0.f,0.f,0.f,0.f,0.f,0.f,0.f,0.f};
#pragma unroll
    for (int kp = 0; kp < 8; ++kp) {
        const v4f b4 = *(const v4f*)(W2p + j * 1024 + kp * 128 + laneoff);
        v2f b0; b0.x = b4.x; b0.y = b4.y;
        v2f b1; b1.x = b4.z; b1.y = b4.w;
        acc = __builtin_amdgcn_wmma_f32_16x16x4_f32(false, a2[2*kp], false, b0,
                                                    (short)0, acc, false, false);
        acc = __builtin_amdgcn_wmma_f32_16x16x4_f32(false, a2[2*kp+1], false, b1,
                                                    (short)0, acc, false, false);
    }
    return acc;
}

__global__ __launch_bounds__(256) void
equi_conv_kernel(const float* __restrict__ pos,
                 const float* __restrict__ f_in,
                 const int*   __restrict__ esrc,
                 const int*   __restrict__ edst,
                 const float* __restrict__ W1p,
                 const float* __restrict__ W2p,
                 float*       __restrict__ fout,
                 int ntiles) {
    __shared__ float embS[WAVES_PB][16][16];  // edges x radial basis
    __shared__ float hS  [WAVES_PB][16][64];  // edges x hidden (post-silu)
    __shared__ float xS  [WAVES_PB][16][40];  // f_in[src] per edge
    __shared__ float dotS[WAVES_PB][16][8];   // x1 . sh1 per (edge,u)
    __shared__ float shS [WAVES_PB][16][4];   // sh1 per edge
    __shared__ int   dstS[WAVES_PB][16];

    const int lane = threadIdx.x & 31;
    const int wv   = threadIdx.x >> 5;
    const int tile = blockIdx.x * WAVES_PB + wv;
    if (tile >= ntiles) return;               // wave-uniform: EXEC stays full

    const int half = lane >> 4;               // 0/1 (K half for WMMA operands)
    const int n    = lane & 15;               // column / local-edge id
    const int e    = n;
    const int ebase   = tile * 16;
    const int laneoff = half * 64 + n * 4;    // packed-B per-lane offset

    // ---------------- Phase A: per-edge geometry, gather, radial basis -----
    const int es = esrc[ebase + e];
    const int ed = edst[ebase + e];
    const float vx = pos[3 * ed + 0] - pos[3 * es + 0];
    const float vy = pos[3 * ed + 1] - pos[3 * es + 1];
    const float vz = pos[3 * ed + 2] - pos[3 * es + 2];
    const float r  = sqrtf(vx * vx + vy * vy + vz * vz + 1e-12f);
    const float sq3_ir = 1.7320508075688772f / r;
    if (half == 0) {
        shS[wv][e][0] = vx * sq3_ir;
        shS[wv][e][1] = vy * sq3_ir;
        shS[wv][e][2] = vz * sq3_ir;
        shS[wv][e][3] = 0.0f;
        dstS[wv][e]   = ed;
    }
    // gather f_in[src]: 20 floats per lane (pair covers 40)
#pragma unroll
    for (int i = 0; i < 20; ++i)
        xS[wv][e][20 * half + i] = f_in[es * 40 + 20 * half + i];

    // radial bump embedding: 8 basis values per lane
    {
        const float step = 5.0f / 17.0f;
        const float coef = 1.14136f * 7.38905609893065f * 4.0f;  // *sqrt(NRB)
#pragma unroll
        for (int i = 0; i < 8; ++i) {
            const int bi = 8 * half + i;
            const float d = (r - step * (float)(bi + 1)) / step;
            float y = 0.0f;
            if (fabsf(d) < 1.0f)
                y = coef * __expf(-1.0f / (1.0f + d)) * __expf(-1.0f / (1.0f - d));
            embS[wv][e][bi] = y;
        }
    }
    // dot[u] = x1[u,:] . sh1  (4 u-values per lane); same-wave DS is in-order
#pragma unroll
    for (int i = 0; i < 4; ++i) {
        const int u = 4 * half + i;
        float s = 0.0f;
#pragma unroll
        for (int k = 0; k < 3; ++k)
            s += xS[wv][e][16 + 3 * u + k] * shS[wv][e][k];
        dotS[wv][e][u] = s;
    }
    asm volatile("s_wait_dscnt 0" ::: "memory");

    // ---------------- GEMM1: emb(16x16) @ W1(16x64) -> silu -> hS ----------
    const int mrow = n;  // A-operand row = lane&15 for both halves
#pragma unroll
    for (int nt = 0; nt < 4; ++nt) {
        v8f acc = {0.f,0.f,0.f,0.f,0.f,0.f,0.f,0.f};
#pragma unroll
        for (int kp = 0; kp < 2; ++kp) {
            const v4f b4 = *(const v4f*)(W1p + nt * 256 + kp * 128 + laneoff);
            v2f b0; b0.x = b4.x; b0.y = b4.y;
            v2f b1; b1.x = b4.z; b1.y = b4.w;
            int kr = 4 * (2 * kp) + 2 * half;
            v2f a0; a0.x = embS[wv][mrow][kr];     a0.y = embS[wv][mrow][kr + 1];
            v2f a1; a1.x = embS[wv][mrow][kr + 4]; a1.y = embS[wv][mrow][kr + 5];
            acc = __builtin_amdgcn_wmma_f32_16x16x4_f32(false, a0, false, b0,
                                                        (short)0, acc, false, false);
            acc = __builtin_amdgcn_wmma_f32_16x16x4_f32(false, a1, false, b1,
                                                        (short)0, acc, false, false);
        }
#pragma unroll
        for (int rr = 0; rr < 8; ++rr) {
            float h = acc[rr] * 0.25f;                 // / sqrt(NRB)
            h = h / (1.0f + __expf(-h));               // silu
            hS[wv][rr + 8 * half][16 * nt + n] = h;
        }
    }
    asm volatile("s_wait_dscnt 0" ::: "memory");

    // A-operands for GEMM2: H as 16(edge) x 64(K)
    v2f a2[16];
#pragma unroll
    for (int kk = 0; kk < 16; ++kk) {
        const int kr = 4 * kk + 2 * half;
        a2[kk].x = hS[wv][mrow][kr];
        a2[kk].y = hS[wv][mrow][kr + 1];
    }

    // ---------------- GEMM2 + fused tensor product -------------------------
    // w tile j covers cols 16j..16j+15; D layout: w[rr] = w[edge rr+8*half][col n]
    v8f acc00  = {0.f,0.f,0.f,0.f,0.f,0.f,0.f,0.f};           // out0[edge][n]
    v8f ac1[3] = {acc00, acc00, acc00};                       // out1[edge][n&7][k]

    for (int j = 0; j < 16; ++j) {            // w00: u = j, wout = n
        v8f w = gemm2_tile(W2p, a2, j, laneoff);
#pragma unroll
        for (int rr = 0; rr < 8; ++rr)
            acc00[rr] += w[rr] * xS[wv][rr + 8 * half][j];
    }
    for (int j = 16; j < 24; ++j) {           // w01: u = 2(j-16)+(n>>3), wout = n&7
        v8f w = gemm2_tile(W2p, a2, j, laneoff);
        const int u = 2 * (j - 16) + (n >> 3);
#pragma unroll
        for (int rr = 0; rr < 8; ++rr) {
            const int ee = rr + 8 * half;
            const float t = w[rr] * xS[wv][ee][u];
            ac1[0][rr] += t * shS[wv][ee][0];
            ac1[1][rr] += t * shS[wv][ee][1];
            ac1[2][rr] += t * shS[wv][ee][2];
        }
    }
    for (int j = 24; j < 28; ++j) {           // w10: u = 2(j-24)+(n>>3), wout = n&7
        v8f w = gemm2_tile(W2p, a2, j, laneoff);
        const int u = 2 * (j - 24) + (n >> 3);
#pragma unroll
        for (int rr = 0; rr < 8; ++rr) {
            const int ee = rr + 8 * half;
            ac1[0][rr] += w[rr] * xS[wv][ee][16 + 3 * u + 0];
            ac1[1][rr] += w[rr] * xS[wv][ee][16 + 3 * u + 1];
            ac1[2][rr] += w[rr] * xS[wv][ee][16 + 3 * u + 2];
        }
    }
    for (int j = 28; j < 36; ++j) {           // w11: u = j-28, wout = n
        v8f w = gemm2_tile(W2p, a2, j, laneoff);
        const int u = j - 28;
#pragma unroll
        for (int rr = 0; rr < 8; ++rr)
            acc00[rr] += INV_S3 * w[rr] * dotS[wv][rr + 8 * half][u];
    }

    // ---------------- Scatter-add into f_out -------------------------------
#pragma unroll
    for (int rr = 0; rr < 8; ++rr) {
        const int dn = dstS[wv][rr + 8 * half];
        atomicAdd(&fout[dn * DIMF + n], acc00[rr] * OUT_SCALE);
    }
#pragma unroll
    for (int k = 0; k < 3; ++k) {
#pragma unroll
        for (int rr = 0; rr < 8; ++rr) {
            const float v = ac1[k][rr];
            const float comb = v + __shfl_xor(v, 8, 32);   // fold n and n+8 dups
            if ((lane & 8) == 0) {
                const int dn = dstS[wv][rr + 8 * half];
                atomicAdd(&fout[dn * DIMF + 16 + 3 * (lane & 7) + k],
                          comb * OUT_SCALE);
            }
        }
    }
}

extern "C" void kernel_launch(void* const* d_in, const int* in_sizes, int n_in,
                              void* d_out, int out_size, void* d_ws, size_t ws_size,
                              hipStream_t stream) {
    const float* pos  = (const float*)d_in[0];
    const float* f_in = (const float*)d_in[1];
    const int*   esrc = (const int*)  d_in[2];
    const int*   edst = (const int*)  d_in[3];
    const float* W1   = (const float*)d_in[4];
    const float* W2   = (const float*)d_in[5];
    float* fout = (float*)d_out;

    float* W2p = (float*)d_ws;                 // 36864 floats
    float* W1p = W2p + W2P_ELEMS;              // 1024 floats

    const int rp_elems = W2P_ELEMS + W1P_ELEMS;
    repack_kernel<<<(rp_elems + 255) / 256, 256, 0, stream>>>(W1, W2, W2p, W1p);
    zero_out_kernel<<<(out_size + 255) / 256, 256, 0, stream>>>(fout, out_size);

    const int nedges = in_sizes[2];
    const int ntiles = (nedges + 15) / 16;                  // 50000
    const int nblocks = (ntiles + WAVES_PB - 1) / WAVES_PB; // 6250
    equi_conv_kernel<<<nblocks, 256, 0, stream>>>(pos, f_in, esrc, edst,
                                                  W1p, W2p, fout, ntiles);
}